// WOMDPostProcessing_52355651338933
// MI455X (gfx1250) — compile-verified
//
#include <hip/hip_runtime.h>
#include <math.h>

// WOMD post-processing for MI455X (gfx1250), wave32.
// One wave per (scenario, agent): n_jf == 32 == wave32 lanes.
// Squared-distance matrix computed DIRECTLY by V_WMMA_F32_16X16X4_F32 using
// K=4 augmentation: A row i = (x_i, y_i, n_i, 1), B col j = (-2x_j, -2y_j, 1, n_j)
//   => D[i][j] = n_i + n_j - 2*e_i.e_j = |e_i - e_j|^2

typedef __attribute__((ext_vector_type(2))) float v2f;
typedef __attribute__((ext_vector_type(8))) float v8f;

#define NSC 64
#define NAG 64
#define NJF 32
#define NST 80
#define KPRED 6

__device__ __forceinline__ float wave_sum(float v) {
#pragma unroll
  for (int o = 16; o; o >>= 1) v += __shfl_xor(v, o, 32);
  return v;
}
__device__ __forceinline__ float wave_max(float v) {
#pragma unroll
  for (int o = 16; o; o >>= 1) v = fmaxf(v, __shfl_xor(v, o, 32));
  return v;
}

__global__ __launch_bounds__(128) void womd_post_kernel(
    const float* __restrict__ ag_type,   // (64, 64, 3)
    const float* __restrict__ trajs,     // (64, 32, 64, 80, 3)
    const float* __restrict__ scores,    // (64, 32, 64)
    float* __restrict__ out_trajs,       // (64, 64, 6, 16, 3)
    float* __restrict__ out_scores)      // (64, 64, 6)
{
  const int lane = threadIdx.x;                       // candidate jf index
  const int sc = blockIdx.x;
  const int ag = blockIdx.y * blockDim.y + threadIdx.y;
  const int l16 = lane & 15;
  const int h = lane >> 4;

  // ---- softmax over the 32 candidate scores (axis = jf) ----
  float s = scores[(sc * NJF + lane) * NAG + ag];
  float m = wave_max(s);
  float p = expf(s - m);
  p /= wave_sum(p);

  // ---- endpoint (last step x,y) per candidate ----
  const float* tb =
      trajs + (((size_t)(sc * NJF + lane) * NAG + ag) * NST + (NST - 1)) * 3;
  float x = tb[0], y = tb[1];
  float nrm = x * x + y * y;

  // ---- per-agent threshold (MTR_THRESH == MPA_THRESH == (2.5,1,2)) ----
  const float* at = ag_type + (sc * NAG + ag) * 3;
  float th = at[0] * 2.5f + at[1] * 1.0f + at[2] * 2.0f;
  float t2 = th * th;

  // ---- WMMA-fused 32x32 squared-distance matrix, four 16x16 tiles ----
  // A 16x4 f32 layout: lanes 0-15 hold K=0,1 of row M=lane; lanes 16-31 hold K=2,3.
  // B 4x16 f32 layout: lanes 0-15 hold K=0,1 of col N=lane; lanes 16-31 hold K=2,3.
  float x0 = __shfl(x, l16, 32),      y0 = __shfl(y, l16, 32);
  float x1 = __shfl(x, 16 + l16, 32), y1 = __shfl(y, 16 + l16, 32);
  float n0 = __shfl(nrm, l16, 32),    n1 = __shfl(nrm, 16 + l16, 32);

  v2f a0, a1, b0, b1;
  a0.x = h ? n0 : x0;           a0.y = h ? 1.0f : y0;          // A block 0
  a1.x = h ? n1 : x1;           a1.y = h ? 1.0f : y1;          // A block 1
  b0.x = h ? 1.0f : -2.0f * x0; b0.y = h ? n0 : -2.0f * y0;    // B block 0
  b1.x = h ? 1.0f : -2.0f * x1; b1.y = h ? n1 : -2.0f * y1;    // B block 1

  v8f zc = {};
  v8f d00 = __builtin_amdgcn_wmma_f32_16x16x4_f32(false, a0, false, b0, (short)0, zc, false, false);
  v8f d01 = __builtin_amdgcn_wmma_f32_16x16x4_f32(false, a0, false, b1, (short)0, zc, false, false);
  v8f d10 = __builtin_amdgcn_wmma_f32_16x16x4_f32(false, a1, false, b0, (short)0, zc, false, false);
  v8f d11 = __builtin_amdgcn_wmma_f32_16x16x4_f32(false, a1, false, b1, (short)0, zc, false, false);

  // D layout: lane holds column N = l16 (+16J), VGPR r holds row M = r + 8h (+16I).
  // Per-lane 32-bit column mask: bit i = (d2(i, j_lane) < t2).
  unsigned pm0 = 0, pm1 = 0;
#pragma unroll
  for (int r = 0; r < 8; ++r) {
    int i0 = r + 8 * h;
    int i1 = 16 + i0;
    if (d00[r] < t2) pm0 |= 1u << i0;
    if (d10[r] < t2) pm0 |= 1u << i1;
    if (d01[r] < t2) pm1 |= 1u << i0;
    if (d11[r] < t2) pm1 |= 1u << i1;
  }
  pm0 |= __shfl_xor(pm0, 16, 32);
  pm1 |= __shfl_xor(pm1, 16, 32);
  unsigned wmask = h ? pm1 : pm0;  // lane j owns within[:, j] (== within[j, :])

  // ---- MTR NMS: 6 rounds of (max-reduce + first-max ballot) + suppression ----
  float cur = p;
  int midx[KPRED];
#pragma unroll
  for (int t = 0; t < KPRED; ++t) {
    float v = wave_max(cur);
    unsigned eq = (unsigned)__ballot(cur == v);
    int bi = __ffs((int)eq) - 1;            // first occurrence == jnp.argmax
    midx[t] = bi;
    bool w = (wmask >> bi) & 1u;
    cur = (lane == bi) ? -1.0f : (w ? cur * 0.01f : cur);
  }

  // ---- lane t (t<6) owns selected mode t ----
  int msel = midx[0];
  msel = (lane == 1) ? midx[1] : msel;
  msel = (lane == 2) ? midx[2] : msel;
  msel = (lane == 3) ? midx[3] : msel;
  msel = (lane == 4) ? midx[4] : msel;
  msel = (lane == 5) ? midx[5] : msel;

  float pl = __shfl(p, msel, 32);
  float s6 = wave_sum(lane < 6 ? pl : 0.0f);
  float qv = pl / s6;  // normalized selected score

  // within2 (6x6) on selected endpoints, same per-agent threshold
  float ex = __shfl(x, msel, 32);
  float ey = __shfl(y, msel, 32);
  unsigned w2 = 0;
#pragma unroll
  for (int b = 0; b < 6; ++b) {
    float bx = __shfl(ex, b, 32), by = __shfl(ey, b, 32);
    float dx = ex - bx, dy = ey - by;
    if (dx * dx + dy * dy < t2) w2 |= 1u << b;
  }

  // stable descending rank of qv among lanes 0..5
  int rank = 0;
#pragma unroll
  for (int b = 0; b < 6; ++b) {
    float qb = __shfl(qv, b, 32);
    if (qb > qv || (qb == qv && b < lane)) rank++;
  }

  // MPA NMS sequential scan in rank order
  float qc = qv;
#pragma unroll
  for (int st = 0; st < 6; ++st) {
    unsigned bal = (unsigned)__ballot(lane < 6 && rank == st);
    int k = __ffs((int)bal) - 1;
    float qk = __shfl(qc, k, 32);
    unsigned wk = (unsigned)__shfl((int)w2, k, 32);
    bool cond = (lane < 6) && ((wk >> lane) & 1u) && (qc > qk);
    bool any = ((unsigned)__ballot(cond)) != 0u;
    if (any && lane == k) qc = 0.001f;
  }

  // normalize + softmax(log(q)/0.5) == q^2 / sum(q^2) (scale-invariant)
  float f = qc * qc;
  float fs = wave_sum(lane < 6 ? f : 0.0f);
  if (lane < 6) out_scores[(sc * NAG + ag) * KPRED + lane] = f / fs;

  // ---- gather trajs[sc, midx[mode], ag, 4:80:5, :] -> out (6,16,3) ----
  float* ot = out_trajs + (size_t)(sc * NAG + ag) * (KPRED * 16 * 3);
#pragma unroll
  for (int it = 0; it < 3; ++it) {
    int mode = it * 2 + h;       // 0..5 across lane halves / iterations
    int stp = l16;               // 0..15
    int jf = h ? midx[it * 2 + 1] : midx[it * 2];
    int srcst = 4 + 5 * stp;
    const float* sp =
        trajs + (((size_t)(sc * NJF + jf) * NAG + ag) * NST + srcst) * 3;
    float a0v = sp[0], a1v = sp[1], a2v = sp[2];
    float* dp = ot + (mode * 16 + stp) * 3;
    dp[0] = a0v; dp[1] = a1v; dp[2] = a2v;
  }
}

extern "C" void kernel_launch(void* const* d_in, const int* in_sizes, int n_in,
                              void* d_out, int out_size, void* d_ws, size_t ws_size,
                              hipStream_t stream) {
  (void)in_sizes; (void)n_in; (void)out_size; (void)d_ws; (void)ws_size;
  const float* ag_type = (const float*)d_in[0];
  const float* trajs   = (const float*)d_in[1];
  const float* scores  = (const float*)d_in[2];
  float* out_trajs  = (float*)d_out;
  float* out_scores = (float*)d_out + (size_t)NSC * NAG * KPRED * 16 * 3;

  dim3 block(32, 4);              // 4 independent waves per block
  dim3 grid(NSC, NAG / 4);        // 4096 (sc, ag) waves total
  hipLaunchKernelGGL(womd_post_kernel, grid, block, 0, stream,
                     ag_type, trajs, scores, out_trajs, out_scores);
}